// VQ_21552145891607
// MI455X (gfx1250) — compile-verified
//
#include <hip/hip_runtime.h>

#define NCODES 512
#define NCH    256
#define NT     16384
#define NBATCH 8
#define CT     (NCH * NT)          // 4,194,304
#define NCT    (NBATCH * CT)       // 33,554,432

#define TB 64                      // time-steps per workgroup
#define XS 260                     // padded LDS row stride (floats): 260 mod 64 = 4 -> conflict-free b64 loads

typedef __attribute__((ext_vector_type(2))) float v2f;
typedef __attribute__((ext_vector_type(8))) float v8f;

// ---------------- kernel 1: dictionary squared norms ----------------
__global__ void vq_wnorm(const float* __restrict__ W, float* __restrict__ wnorm) {
    int d = blockIdx.x * blockDim.x + threadIdx.x;
    if (d < NCODES) {
        const float* row = W + d * NCH;
        float s = 0.f;
#pragma unroll 8
        for (int c = 0; c < NCH; ++c) s += row[c] * row[c];
        wnorm[d] = s;
    }
}

// ---------------- kernel 2: pack W into WMMA A-operand lane order ----------------
// A (16x4 f32): VGPR0 = {K=0 lanes0-15, K=2 lanes16-31}, VGPR1 = {K=1, K=3}
// Layout: [dtg][ks][chain j][lane] so the 4 chains' A operands per K-step are
// one contiguous 1KB block (clean 16-load clauses in the main loop).
__global__ void vq_pack(const float* __restrict__ W, v2f* __restrict__ Ap) {
    int t = blockIdx.x * blockDim.x + threadIdx.x;   // 65536 threads
    int lane = t & 31;
    int j    = (t >> 5) & 3;     // chain within dt-group
    int ks   = (t >> 7) & 63;    // K-step
    int dtg  = t >> 13;          // dt-group of 4 code tiles
    int dt   = dtg * 4 + j;
    int m = lane & 15, h = lane >> 4;
    int row = dt * 16 + m;
    int col = 4 * ks + 2 * h;
    v2f a;
    a.x = W[row * NCH + col];
    a.y = W[row * NCH + col + 1];
    Ap[t] = a;
}

// ---------------- kernel 3: main VQ ----------------
__global__ void vq_main(const float* __restrict__ in, const float* __restrict__ W,
                        const float* __restrict__ wnormG, const v2f* __restrict__ Ap,
                        float* __restrict__ out1, float* __restrict__ out2,
                        int* __restrict__ idxOut) {
    extern __shared__ float smem[];
    float* xtile = smem;                    // [TB][XS] input tile, [b][c] layout
    float* wn    = smem + TB * XS;          // [NCODES]
    int*   idxs  = (int*)(wn + NCODES);     // [TB]

    const int tid = threadIdx.x;            // 128 threads = 4 waves
    const int n   = blockIdx.y;
    const int t0  = blockIdx.x * TB;
    const float* xin = in + n * CT + t0;

    // ---- stage x tile: global [c][t contiguous] -> LDS [b][c] (transpose) ----
#if __has_builtin(__builtin_amdgcn_global_load_async_to_lds_b32) && \
    __has_builtin(__builtin_amdgcn_s_wait_asynccnt)
    {
        // Async per-lane scatter into LDS: tracked by ASYNCcnt, no VGPR landing.
        const int b     = tid & 63;
        const int c_off = tid >> 6;          // 0..1
        __attribute__((address_space(3))) int* lbase =
            (__attribute__((address_space(3))) int*)xtile;
        for (int cb = 0; cb < NCH; cb += 2) {
            int c = cb + c_off;
            float* g = const_cast<float*>(xin + c * NT + b);
            __builtin_amdgcn_global_load_async_to_lds_b32(
                (__attribute__((address_space(1))) int*)g,
                lbase + (b * XS + c),
                0, 0);
        }
        __builtin_amdgcn_s_wait_asynccnt(0);
    }
#else
    {
        const int c_off = tid >> 4;          // 0..7
        const int bq    = (tid & 15) * 4;    // 0..60
        for (int cb = 0; cb < NCH; cb += 8) {
            int c = cb + c_off;
            const float4 v = *(const float4*)(xin + c * NT + bq);
            xtile[(bq + 0) * XS + c] = v.x;
            xtile[(bq + 1) * XS + c] = v.y;
            xtile[(bq + 2) * XS + c] = v.z;
            xtile[(bq + 3) * XS + c] = v.w;
        }
    }
#endif
    for (int i = tid; i < NCODES; i += 128) wn[i] = wnormG[i];
    __syncthreads();

    const int wave = tid >> 5;
    const int lane = tid & 31;
    const int bl   = (wave << 4) + (lane & 15);  // local time-step 0..63
    const int kh   = lane >> 4;                  // lane half selects K-pair / M-half

    const float* xrow = xtile + bl * XS + 2 * kh;

    float bestS = 3.4e38f;
    int   bestI = 0;

    // sweep code tiles in groups of 4: 4 independent accumulator chains
    // (dependency distance 4) sharing one B operand per K-step.
    for (int dtg = 0; dtg < 8; ++dtg) {
        v8f acc0 = {}, acc1 = {}, acc2 = {}, acc3 = {};
        const v2f* A = Ap + dtg * 8192 + lane;
#pragma unroll 2
        for (int ks = 0; ks < 64; ++ks) {
            v2f b  = *(const v2f*)(xrow + 4 * ks);   // ds_load_b64, conflict-free
            v2f a0 = A[(ks * 4 + 0) * 32];           // contiguous 1KB clause
            v2f a1 = A[(ks * 4 + 1) * 32];
            v2f a2 = A[(ks * 4 + 2) * 32];
            v2f a3 = A[(ks * 4 + 3) * 32];
            acc0 = __builtin_amdgcn_wmma_f32_16x16x4_f32(
                       false, a0, false, b, (short)0, acc0, false, false);
            acc1 = __builtin_amdgcn_wmma_f32_16x16x4_f32(
                       false, a1, false, b, (short)0, acc1, false, false);
            acc2 = __builtin_amdgcn_wmma_f32_16x16x4_f32(
                       false, a2, false, b, (short)0, acc2, false, false);
            acc3 = __builtin_amdgcn_wmma_f32_16x16x4_f32(
                       false, a3, false, b, (short)0, acc3, false, false);
        }
        // D[m,n]: VGPR r -> code d = (dtg*4 + j)*16 + r + 8*kh ; lane%16 -> time-step
        const int dbase = dtg * 64 + 8 * kh;
#pragma unroll
        for (int r = 0; r < 8; ++r) {
            float s = wn[dbase + r] - 2.0f * acc0[r];
            if (s < bestS) { bestS = s; bestI = dbase + r; }
        }
#pragma unroll
        for (int r = 0; r < 8; ++r) {
            float s = wn[dbase + 16 + r] - 2.0f * acc1[r];
            if (s < bestS) { bestS = s; bestI = dbase + 16 + r; }
        }
#pragma unroll
        for (int r = 0; r < 8; ++r) {
            float s = wn[dbase + 32 + r] - 2.0f * acc2[r];
            if (s < bestS) { bestS = s; bestI = dbase + 32 + r; }
        }
#pragma unroll
        for (int r = 0; r < 8; ++r) {
            float s = wn[dbase + 48 + r] - 2.0f * acc3[r];
            if (s < bestS) { bestS = s; bestI = dbase + 48 + r; }
        }
    }

    // combine the two lane-halves holding the same time-step (wave32 xor-16)
    {
        float oS = __shfl_xor(bestS, 16, 32);
        int   oI = __shfl_xor(bestI, 16, 32);
        if (oS < bestS || (oS == bestS && oI < bestI)) { bestS = oS; bestI = oI; }
    }
    if (kh == 0) {
        idxs[bl] = bestI;
        idxOut[n * NT + t0 + bl] = bestI;
    }
    __syncthreads();

    // ---- cooperative embedded write: gather 4 dictionary rows per thread,
    // ---- store b128 coalesced along t (bandwidth-bound tail, fewest stores) ----
    {
        const int q  = (tid & 15) * 4;           // time-step quad base
        const int c0 = tid >> 4;                 // 0..7
        const float* w0 = W + idxs[q + 0] * NCH;
        const float* w1 = W + idxs[q + 1] * NCH;
        const float* w2 = W + idxs[q + 2] * NCH;
        const float* w3 = W + idxs[q + 3] * NCH;
        const int obase = n * CT + t0 + q;
#pragma unroll 4
        for (int c = c0; c < NCH; c += 8) {
            float4 v;
            v.x = w0[c]; v.y = w1[c]; v.z = w2[c]; v.w = w3[c];
            *(float4*)(out1 + obase + c * NT) = v;
            *(float4*)(out2 + obase + c * NT) = v;   // straight-through is bit-identical
        }
    }
}

extern "C" void kernel_launch(void* const* d_in, const int* in_sizes, int n_in,
                              void* d_out, int out_size, void* d_ws, size_t ws_size,
                              hipStream_t stream) {
    const float* in = (const float*)d_in[0];
    const float* W  = (const float*)d_in[1];

    float* out1  = (float*)d_out;
    float* out2  = out1 + (size_t)NCT;
    int*   idxO  = (int*)(out1 + 2 * (size_t)NCT);

    float* wnorm = (float*)d_ws;                       // 512 floats
    v2f*   Ap    = (v2f*)((float*)d_ws + NCODES);      // 512 KB packed A operand

    vq_wnorm<<<dim3(2), dim3(256), 0, stream>>>(W, wnorm);
    vq_pack<<<dim3(65536 / 256), dim3(256), 0, stream>>>(W, Ap);

    const size_t ldsBytes = (size_t)(TB * XS + NCODES) * sizeof(float) + TB * sizeof(int);
    vq_main<<<dim3(NT / TB, NBATCH), dim3(128), ldsBytes, stream>>>(
        in, W, wnorm, Ap, out1, out2, idxO);
}